// Group_37726992728311
// MI455X (gfx1250) — compile-verified
//
#include <hip/hip_runtime.h>
#include <hip/hip_bf16.h>

// Problem constants (match reference setup_inputs)
#define BB  16
#define NN  16384
#define GG  512       // NUM_GROUP
#define KK  32        // GROUP_SIZE
#define NT  (NN / 16) // 1024 16-point tiles

typedef float v2f __attribute__((ext_vector_type(2)));
typedef float v8f __attribute__((ext_vector_type(8)));

// ---------------------------------------------------------------------------
// Kernel 1: farthest point sampling. One block per batch, 1024 threads.
// Running min-distance kept in 16 registers per thread; argmax via wave
// shuffle + LDS cross-wave reduce. Writes centers as (x,y,z,1) float4 for the
// WMMA A-operand and the raw center xyz into the output tail.
// ---------------------------------------------------------------------------
__global__ __launch_bounds__(1024) void fps_kernel(const float* __restrict__ xyz,
                                                   float* __restrict__ cen4,
                                                   float* __restrict__ outCen) {
  const int b    = blockIdx.x;
  const int tid  = threadIdx.x;
  const int lane = tid & 31;
  const int wid  = tid >> 5;
  const float* p = xyz + (size_t)b * NN * 3;

  __shared__ float rsum[96];
  __shared__ float rval[32];
  __shared__ int   rind[32];
  __shared__ float sbar[3];
  __shared__ int   sfar;

  // --- barycenter ---
  float sx = 0.f, sy = 0.f, sz = 0.f;
  for (int j = 0; j < 16; ++j) {
    int n = tid + j * 1024;
    sx += p[n * 3 + 0]; sy += p[n * 3 + 1]; sz += p[n * 3 + 2];
  }
  for (int off = 16; off > 0; off >>= 1) {
    sx += __shfl_down(sx, off, 32);
    sy += __shfl_down(sy, off, 32);
    sz += __shfl_down(sz, off, 32);
  }
  if (lane == 0) { rsum[wid] = sx; rsum[32 + wid] = sy; rsum[64 + wid] = sz; }
  __syncthreads();
  if (tid == 0) {
    float ax = 0.f, ay = 0.f, az = 0.f;
    for (int w = 0; w < 32; ++w) { ax += rsum[w]; ay += rsum[32 + w]; az += rsum[64 + w]; }
    sbar[0] = ax / (float)NN; sbar[1] = ay / (float)NN; sbar[2] = az / (float)NN;
  }
  __syncthreads();

  // --- initial farthest = argmax dist to barycenter ---
  const float bx = sbar[0], by = sbar[1], bz = sbar[2];
  float dist[16];
  float bestv = -1.f; int besti = 0;
  for (int j = 0; j < 16; ++j) {
    int n = tid + j * 1024;
    float dx = p[n * 3 + 0] - bx, dy = p[n * 3 + 1] - by, dz = p[n * 3 + 2] - bz;
    float d = dx * dx + dy * dy + dz * dz;
    dist[j] = 1e10f;
    if (d > bestv) { bestv = d; besti = n; }
  }
  for (int off = 16; off > 0; off >>= 1) {
    float ov = __shfl_down(bestv, off, 32);
    int   oi = __shfl_down(besti, off, 32);
    if (ov > bestv || (ov == bestv && oi < besti)) { bestv = ov; besti = oi; }
  }
  if (lane == 0) { rval[wid] = bestv; rind[wid] = besti; }
  __syncthreads();
  if (tid == 0) {
    float bv = rval[0]; int bi = rind[0];
    for (int w = 1; w < 32; ++w)
      if (rval[w] > bv || (rval[w] == bv && rind[w] < bi)) { bv = rval[w]; bi = rind[w]; }
    sfar = bi;
  }
  __syncthreads();
  int farthest = sfar;

  // --- 512 sequential FPS steps ---
  for (int it = 0; it < GG; ++it) {
    // centroid = xyz[farthest]; same-address loads broadcast from L2/WGP$
    float cx = p[farthest * 3 + 0];
    float cy = p[farthest * 3 + 1];
    float cz = p[farthest * 3 + 2];
    if (tid == 0) {
      size_t gi = (size_t)b * GG + it;
      cen4[gi * 4 + 0] = cx; cen4[gi * 4 + 1] = cy;
      cen4[gi * 4 + 2] = cz; cen4[gi * 4 + 3] = 1.0f;
      outCen[gi * 3 + 0] = cx; outCen[gi * 3 + 1] = cy; outCen[gi * 3 + 2] = cz;
    }
    bestv = -1.f; besti = 0;
    for (int j = 0; j < 16; ++j) {
      int n = tid + j * 1024;
      float dx = p[n * 3 + 0] - cx, dy = p[n * 3 + 1] - cy, dz = p[n * 3 + 2] - cz;
      float d = dx * dx + dy * dy + dz * dz;
      dist[j] = fminf(dist[j], d);
      if (dist[j] > bestv) { bestv = dist[j]; besti = n; }
    }
    for (int off = 16; off > 0; off >>= 1) {
      float ov = __shfl_down(bestv, off, 32);
      int   oi = __shfl_down(besti, off, 32);
      if (ov > bestv || (ov == bestv && oi < besti)) { bestv = ov; besti = oi; }
    }
    if (lane == 0) { rval[wid] = bestv; rind[wid] = besti; }
    __syncthreads();
    if (tid == 0) {
      float bv = rval[0]; int bi = rind[0];
      for (int w = 1; w < 32; ++w)
        if (rval[w] > bv || (rval[w] == bv && rind[w] < bi)) { bv = rval[w]; bi = rind[w]; }
      sfar = bi;
    }
    __syncthreads();
    farthest = sfar;
  }
}

// ---------------------------------------------------------------------------
// Kernel 2: build WMMA B-operand table: per point (-2x,-2y,-2z,|x|^2) so one
// 16x16x4 f32 WMMA tile produces score s = |x|^2 - 2 c.x (same ordering as d2,
// since the per-center |c|^2 offset cannot change neighbor ranking).
// ---------------------------------------------------------------------------
__global__ void prep_kernel(const float* __restrict__ xyz, float* __restrict__ pt4) {
  int i = blockIdx.x * blockDim.x + threadIdx.x;
  if (i < BB * NN) {
    float x = xyz[i * 3 + 0], y = xyz[i * 3 + 1], z = xyz[i * 3 + 2];
    pt4[i * 4 + 0] = -2.0f * x;
    pt4[i * 4 + 1] = -2.0f * y;
    pt4[i * 4 + 2] = -2.0f * z;
    pt4[i * 4 + 3] = x * x + y * y + z * z;
  }
}

// ---------------------------------------------------------------------------
// Kernel 3: KNN via V_WMMA_F32_16X16X4_F32 + streaming top-32 + gather.
// One wave per group of 16 centers: grid (G/16, B) = (32, 16).
// Software-pipelined: B operands for two tiles are preloaded and rotated so
// the next pair of global_load_b64 overlaps the LDS score writes, the
// selection scan and the barrier of the current pair. Two back-to-back WMMAs
// per iteration halve the barrier count.
// ---------------------------------------------------------------------------
__global__ __launch_bounds__(32) void knn_kernel(const float* __restrict__ xyz,
                                                 const float* __restrict__ pt4,
                                                 const float* __restrict__ cen4,
                                                 float* __restrict__ outNbr) {
  const int b  = blockIdx.y;
  const int g0 = blockIdx.x * 16;
  const int l  = threadIdx.x;     // 0..31
  const int col = l & 15;         // M for A-load, N for D-tile
  const int hi  = l >> 4;         // lane half selects K pair (0,1) vs (2,3)

  __shared__ float sScore[512];       // two 16x16 score tiles
  __shared__ float sTopD[16][KK];
  __shared__ int   sTopI[16][KK];

  for (int e = l; e < 16 * KK; e += 32) {
    (&sTopD[0][0])[e] = 3.0e38f;
    (&sTopI[0][0])[e] = 0;
  }
  __syncthreads();

  // A operand: lane layout per ISA (16x4 f32 A): vgpr0 = K{0|2}, vgpr1 = K{1|3}
  const float* ca = cen4 + ((size_t)(b * GG + g0 + col)) * 4 + hi * 2;
  v2f Aop; Aop.x = ca[0]; Aop.y = ca[1];

  // B operand base for this lane: tile t lives at +t*64 floats
  const float* pbase = pt4 + (size_t)b * NN * 4 + (size_t)col * 4 + hi * 2;

  float thresh = 3.0e38f;
  int   maxpos = 0;

  // prologue: preload tiles 0 and 1
  v2f Bc0, Bc1;
  Bc0.x = pbase[0];  Bc0.y = pbase[1];
  Bc1.x = pbase[64]; Bc1.y = pbase[65];

  for (int t = 0; t < NT; t += 2) {
    v8f a0 = {}, a1 = {};
    a0 = __builtin_amdgcn_wmma_f32_16x16x4_f32(
        false, Aop, false, Bc0, (short)0, a0, false, false);
    a1 = __builtin_amdgcn_wmma_f32_16x16x4_f32(
        false, Aop, false, Bc1, (short)0, a1, false, false);

    // rotate: issue next pair's loads now; they complete under the score
    // writes + selection scan + barrier below.
    if (t + 2 < NT) {
      const float* np = pbase + (size_t)(t + 2) * 64;
      Bc0.x = np[0];  Bc0.y = np[1];
      Bc1.x = np[64]; Bc1.y = np[65];
      __builtin_prefetch(np + 256, 0, 3);   // ~4 tiles ahead, near-cache scope
    }

#pragma unroll
    for (int v = 0; v < 8; ++v) {
      int row = v + hi * 8;                    // D layout: M = vgpr + 8*(lane>=16)
      sScore[row * 16 + col]       = a0[v];    // N = lane & 15
      sScore[256 + row * 16 + col] = a1[v];
    }
    __syncthreads();

    if (l < 16) {                              // lane l owns center row l
      for (int j = 0; j < 32; ++j) {
        float s = sScore[(j >> 4) * 256 + l * 16 + (j & 15)];
        if (s < thresh) {
          sTopD[l][maxpos] = s;
          sTopI[l][maxpos] = t * 16 + j;       // j>=16 falls into tile t+1
          float mx = -3.0e38f; int mp = 0;
          for (int q = 0; q < KK; ++q) {
            float dq = sTopD[l][q];
            if (dq > mx) { mx = dq; mp = q; }
          }
          thresh = mx; maxpos = mp;
        }
      }
    }
    __syncthreads();                           // EXEC reconverged before next WMMA
  }

  // sort each owner's 32 entries ascending by score (ties: lower index first)
  if (l < 16) {
    for (int a = 0; a < KK - 1; ++a) {
      int best = a; float bv = sTopD[l][a];
      for (int q = a + 1; q < KK; ++q) {
        float dq = sTopD[l][q];
        if (dq < bv || (dq == bv && sTopI[l][q] < sTopI[l][best])) { bv = dq; best = q; }
      }
      if (best != a) {
        float td = sTopD[l][a]; sTopD[l][a] = sTopD[l][best]; sTopD[l][best] = td;
        int   ti = sTopI[l][a]; sTopI[l][a] = sTopI[l][best]; sTopI[l][best] = ti;
      }
    }
  }
  __syncthreads();

  // gather neighborhoods and subtract center
  const float* px = xyz + (size_t)b * NN * 3;
  for (int e = l; e < 16 * KK; e += 32) {
    int g = e >> 5;                      // center within this tile
    int k = e & 31;
    int idx = sTopI[g][k];
    const float* cc = cen4 + ((size_t)(b * GG + g0 + g)) * 4;
    size_t o = ((size_t)(b * GG + g0 + g) * KK + k) * 3;
    outNbr[o + 0] = px[idx * 3 + 0] - cc[0];
    outNbr[o + 1] = px[idx * 3 + 1] - cc[1];
    outNbr[o + 2] = px[idx * 3 + 2] - cc[2];
  }
}

// ---------------------------------------------------------------------------
extern "C" void kernel_launch(void* const* d_in, const int* in_sizes, int n_in,
                              void* d_out, int out_size, void* d_ws, size_t ws_size,
                              hipStream_t stream) {
  (void)in_sizes; (void)n_in; (void)out_size; (void)ws_size;
  const float* xyz = (const float*)d_in[0];

  float* outNbr = (float*)d_out;                       // [16][512][32][3]
  float* outCen = outNbr + (size_t)BB * GG * KK * 3;   // [16][512][3]

  float* pt4  = (float*)d_ws;                          // B*N*4 floats (4 MB)
  float* cen4 = pt4 + (size_t)BB * NN * 4;             // B*G*4 floats (128 KB)

  fps_kernel<<<BB, 1024, 0, stream>>>(xyz, cen4, outCen);
  prep_kernel<<<(BB * NN + 255) / 256, 256, 0, stream>>>(xyz, pt4);
  knn_kernel<<<dim3(GG / 16, BB), 32, 0, stream>>>(xyz, pt4, cen4, outNbr);
}